// SelfAttention_64218351010503
// MI455X (gfx1250) — compile-verified
//
#include <hip/hip_runtime.h>

#define BATCH 2
#define CCH   128
#define NPTS  8192
#define KNN   16
#define EPSV  1e-5f
#define SLOPE 0.2f

typedef __attribute__((ext_vector_type(2))) float v2f;
typedef __attribute__((ext_vector_type(8))) float v8f;

__device__ __forceinline__ v8f wmma_f32_4(v2f a, v2f b, v8f c) {
  // D = A(16x4) * B(4x16) + C(16x16), fp32 throughout.
  return __builtin_amdgcn_wmma_f32_16x16x4_f32(false, a, false, b, (short)0, c, false, false);
}

// --------------------------- async global -> LDS ---------------------------
#if defined(__has_builtin)
#if __has_builtin(__builtin_amdgcn_global_load_async_to_lds_b128)
#define USE_ASYNC_LDS 1
#endif
#endif
#ifndef USE_ASYNC_LDS
#define USE_ASYNC_LDS 0
#endif

#if USE_ASYNC_LDS
typedef int v4i_vs __attribute__((vector_size(16)));   // matches builtin param pointee
__device__ __forceinline__ void async_copy16(const float* g, float* l) {
  __builtin_amdgcn_global_load_async_to_lds_b128((v4i_vs*)g, (v4i_vs*)l, 0, 0);
}
__device__ __forceinline__ void async_wait0() {
#if __has_builtin(__builtin_amdgcn_s_wait_asynccnt)
  __builtin_amdgcn_s_wait_asynccnt(0);
#else
  asm volatile("s_wait_asynccnt 0x0" ::: "memory");
#endif
}
#endif

// ---------------------------------------------------------------------------
// features [B,C,N] -> xT [B,N,C]  (so a point's feature vector is contiguous)
// ---------------------------------------------------------------------------
__global__ __launch_bounds__(256) void transpose_feat_kernel(
    const float* __restrict__ f, float* __restrict__ fT) {
  const int t = blockIdx.x * 256 + threadIdx.x;   // B*N*128 = 2^21
  const int c = t & 127;
  const int n = (t >> 7) & (NPTS - 1);
  const int b = t >> 20;
  fT[t] = f[((size_t)b * CCH + c) * NPTS + n];
}

// ---------------------------------------------------------------------------
// KNN: one thread per point, tiled scan over all points, register top-16.
// ---------------------------------------------------------------------------
__global__ __launch_bounds__(256) void knn_kernel(
    const float* __restrict__ coords, int* __restrict__ nidx) {
  const int gid = blockIdx.x * 256 + threadIdx.x; // 0 .. B*N-1
  const int b = gid / NPTS;
  const int n = gid & (NPTS - 1);
  const float* cb = coords + (size_t)b * 3 * NPTS;
  const float px = cb[n];
  const float py = cb[NPTS + n];
  const float pz = cb[2 * NPTS + n];

  float bestd[KNN];
  int   besti[KNN];
#pragma unroll
  for (int i = 0; i < KNN; ++i) { bestd[i] = 3.4e38f; besti[i] = 0; }

  __shared__ float sx[256], sy[256], sz[256];
  for (int t0 = 0; t0 < NPTS; t0 += 256) {
    __syncthreads();
    const int m = t0 + threadIdx.x;
    sx[threadIdx.x] = cb[m];
    sy[threadIdx.x] = cb[NPTS + m];
    sz[threadIdx.x] = cb[2 * NPTS + m];
    __syncthreads();
    for (int j = 0; j < 256; ++j) {
      const int m2 = t0 + j;
      const float dx = px - sx[j], dy = py - sy[j], dz = pz - sz[j];
      const float d2 = dx * dx + dy * dy + dz * dz;
      if (d2 < bestd[KNN - 1] && m2 != n) {
        float pd = d2; int pi = m2;
#pragma unroll
        for (int i = 0; i < KNN; ++i) {       // bubble-insert, static indices only
          const bool sw = pd < bestd[i];
          const float td = bestd[i]; const int ti = besti[i];
          bestd[i] = sw ? pd : td;  besti[i] = sw ? pi : ti;
          pd       = sw ? td : pd;  pi       = sw ? ti : pi;
        }
      }
    }
  }
  int* ob = nidx + ((size_t)gid << 4);
#pragma unroll
  for (int k = 0; k < KNN; ++k) ob[k] = besti[k];
}

// ---------------------------------------------------------------------------
// EdgeConv GEMM: y[o,(n,k)] = W[o,:] . [center ; neigh-center]
// Wave tile = 16 o x 16 k (one point). Produces max_k(y) and sum/sumsq stats.
// ---------------------------------------------------------------------------
template <int OCH>
__global__ __launch_bounds__(256) void edge_gemm_kernel(
    const float* __restrict__ xT,      // [B,N,128]
    const float* __restrict__ W,       // [OCH,256]
    const int*   __restrict__ nidx,    // [B,N,16]
    float* __restrict__ ymaxT,         // [B,N,OCH]
    float* __restrict__ ssum, float* __restrict__ ssq) {
  constexpr int NCHUNK = 64;
  constexpr int CHPTS  = NPTS / NCHUNK;   // 128 points per block
  constexpr int REPS   = OCH / 128;
  constexpr int NSTR   = CCH + 4;         // 132: 16B-aligned rows, conflict-free reads
  __shared__ __align__(16) float s_center[CCH];
  __shared__ __align__(16) float s_neigh[KNN * NSTR];

  const int t = threadIdx.x;
  const int wave = t >> 5, lane = t & 31, half = lane >> 4, l15 = lane & 15;
  const int b = blockIdx.x / NCHUNK;
  const int n_start = (blockIdx.x % NCHUNK) * CHPTS;

  float sacc[REPS][8], qacc[REPS][8];
#pragma unroll
  for (int rp = 0; rp < REPS; ++rp)
#pragma unroll
    for (int r = 0; r < 8; ++r) { sacc[rp][r] = 0.f; qacc[rp][r] = 0.f; }

  for (int p = 0; p < CHPTS; ++p) {
    const int n = n_start + p;
    __syncthreads();
    const int* ib = nidx + ((size_t)(b * NPTS + n) << 4);
#if USE_ASYNC_LDS
    // 16 neighbors x 128ch staged as 512 x b128 async transfers + 32 for center
    if (t < 32)
      async_copy16(xT + ((size_t)b * NPTS + n) * CCH + t * 4, &s_center[t * 4]);
#pragma unroll
    for (int e = t; e < KNN * CCH / 4; e += 256) {
      const int k  = e >> 5;
      const int c4 = (e & 31) << 2;
      const int nb = ib[k];
      async_copy16(xT + ((size_t)b * NPTS + nb) * CCH + c4, &s_neigh[k * NSTR + c4]);
    }
    async_wait0();
#else
    if (t < CCH) s_center[t] = xT[((size_t)b * NPTS + n) * CCH + t];
    for (int e = t; e < KNN * CCH; e += 256) {
      const int k = e >> 7;
      const int c = e & (CCH - 1);
      const int nb = ib[k];
      s_neigh[k * NSTR + c] = xT[((size_t)b * NPTS + nb) * CCH + c];
    }
#endif
    __syncthreads();

#pragma unroll
    for (int rp = 0; rp < REPS; ++rp) {
      const int o0 = (wave + rp * 8) * 16;
      const float* Wrow = W + (size_t)(o0 + l15) * 256;
      v8f acc = {};
      // center half of channels
#pragma unroll 8
      for (int c0 = 0; c0 < 128; c0 += 4) {
        const int cA = c0 + half * 2;
        v2f a  = { Wrow[cA], Wrow[cA + 1] };
        v2f bb = { s_center[cA], s_center[cA + 1] };
        acc = wmma_f32_4(a, bb, acc);
      }
      // (neighbor - center) half
      const float* s_nb = s_neigh + l15 * NSTR;
#pragma unroll 8
      for (int c0 = 0; c0 < 128; c0 += 4) {
        const int cA = c0 + half * 2;
        v2f a  = { Wrow[128 + cA], Wrow[128 + cA + 1] };
        v2f bb = { s_nb[cA] - s_center[cA], s_nb[cA + 1] - s_center[cA + 1] };
        acc = wmma_f32_4(a, bb, acc);
      }
      // stats + max over k (16 lanes of each half hold k=0..15 for one o)
#pragma unroll
      for (int r = 0; r < 8; ++r) {
        const float v = acc[r];
        sacc[rp][r] += v;
        qacc[rp][r] += v * v;
        float mx = v;
        mx = fmaxf(mx, __shfl_xor(mx, 1));
        mx = fmaxf(mx, __shfl_xor(mx, 2));
        mx = fmaxf(mx, __shfl_xor(mx, 4));
        mx = fmaxf(mx, __shfl_xor(mx, 8));
        if (l15 == 0) {
          const int o = o0 + r + half * 8;
          ymaxT[((size_t)b * NPTS + n) * OCH + o] = mx;
        }
      }
    }
  }
  // flush per-o partial sums (one atomic per (wave,r,half) per block)
#pragma unroll
  for (int rp = 0; rp < REPS; ++rp)
#pragma unroll
    for (int r = 0; r < 8; ++r) {
      float s = sacc[rp][r], q = qacc[rp][r];
      s += __shfl_xor(s, 1); s += __shfl_xor(s, 2); s += __shfl_xor(s, 4); s += __shfl_xor(s, 8);
      q += __shfl_xor(q, 1); q += __shfl_xor(q, 2); q += __shfl_xor(q, 4); q += __shfl_xor(q, 8);
      if (l15 == 0) {
        const int o = (wave + rp * 8) * 16 + r + half * 8;
        atomicAdd(&ssum[b * OCH + o], s);
        atomicAdd(&ssq[b * OCH + o], q);
      }
    }
}

// ---------------------------------------------------------------------------
// Final conv: y3[o,n] = W3[o,:] . concat(x0,x1,x2)[.,n]; columns = 16 points.
// ---------------------------------------------------------------------------
__global__ __launch_bounds__(256) void final_gemm_kernel(
    const float* __restrict__ x0T, const float* __restrict__ x1T,
    const float* __restrict__ x2T, const float* __restrict__ W3,
    float* __restrict__ y3, float* __restrict__ ssum, float* __restrict__ ssq) {
  constexpr int CIN = 512;
  constexpr int LSTR = CIN + 4;     // 516: 16B-aligned rows, conflict-free reads
  constexpr int GROUPS = 4;
  constexpr int BLOCKS_PER_B = NPTS / (GROUPS * 16);   // 128
  __shared__ __align__(16) float s_x[16 * LSTR];

  const int t = threadIdx.x;
  const int wave = t >> 5, lane = t & 31, half = lane >> 4, l15 = lane & 15;
  const int b = blockIdx.x / BLOCKS_PER_B;
  const int n_base = (blockIdx.x % BLOCKS_PER_B) * (GROUPS * 16);
  const float* Wrow = W3 + (size_t)(wave * 16 + l15) * CIN;

  float sacc[8] = {0, 0, 0, 0, 0, 0, 0, 0};
  float qacc[8] = {0, 0, 0, 0, 0, 0, 0, 0};

  for (int g = 0; g < GROUPS; ++g) {
    const int n0 = n_base + g * 16;
    __syncthreads();
#if USE_ASYNC_LDS
#pragma unroll
    for (int e = t; e < 16 * CIN / 4; e += 256) {   // 2048 x b128
      const int row = e >> 7;
      const int c4  = (e & 127) << 2;
      const int n = n0 + row;
      const float* src;
      if (c4 < 128)      src = x0T + ((size_t)b * NPTS + n) * 128 + c4;
      else if (c4 < 256) src = x1T + ((size_t)b * NPTS + n) * 128 + (c4 - 128);
      else               src = x2T + ((size_t)b * NPTS + n) * 256 + (c4 - 256);
      async_copy16(src, &s_x[row * LSTR + c4]);
    }
    async_wait0();
#else
    for (int e = t; e < 16 * CIN; e += 256) {
      const int row = e >> 9;
      const int c = e & (CIN - 1);
      const int n = n0 + row;
      float v;
      if (c < 128)      v = x0T[((size_t)b * NPTS + n) * 128 + c];
      else if (c < 256) v = x1T[((size_t)b * NPTS + n) * 128 + (c - 128)];
      else              v = x2T[((size_t)b * NPTS + n) * 256 + (c - 256)];
      s_x[row * LSTR + c] = v;
    }
#endif
    __syncthreads();
    v8f acc = {};
    const float* srow = s_x + l15 * LSTR;
#pragma unroll 8
    for (int c0 = 0; c0 < CIN; c0 += 4) {
      const int cA = c0 + half * 2;
      v2f a  = { Wrow[cA], Wrow[cA + 1] };
      v2f bb = { srow[cA], srow[cA + 1] };
      acc = wmma_f32_4(a, bb, acc);
    }
#pragma unroll
    for (int r = 0; r < 8; ++r) {
      const float v = acc[r];
      sacc[r] += v;
      qacc[r] += v * v;
      const int o = wave * 16 + r + half * 8;
      y3[((size_t)b * 128 + o) * NPTS + n0 + l15] = v;
    }
  }
#pragma unroll
  for (int r = 0; r < 8; ++r) {
    float s = sacc[r], q = qacc[r];
    s += __shfl_xor(s, 1); s += __shfl_xor(s, 2); s += __shfl_xor(s, 4); s += __shfl_xor(s, 8);
    q += __shfl_xor(q, 1); q += __shfl_xor(q, 2); q += __shfl_xor(q, 4); q += __shfl_xor(q, 8);
    if (l15 == 0) {
      const int o = wave * 16 + r + half * 8;
      atomicAdd(&ssum[b * 128 + o], s);
      atomicAdd(&ssq[b * 128 + o], q);
    }
  }
}

// ---------------------------------------------------------------------------
__global__ void finalize_stats_kernel(const float* __restrict__ ssum,
                                      const float* __restrict__ ssq,
                                      float* __restrict__ mean, float* __restrict__ rstd,
                                      int total, float inv_cnt) {
  const int i = blockIdx.x * blockDim.x + threadIdx.x;
  if (i < total) {
    const float m = ssum[i] * inv_cnt;
    const float v = ssq[i] * inv_cnt - m * m;
    mean[i] = m;
    rstd[i] = rsqrtf(fmaxf(v, 0.f) + EPSV);
  }
}

template <int OCH>
__global__ __launch_bounds__(256) void apply_norm_T_kernel(
    const float* __restrict__ in, const float* __restrict__ mean,
    const float* __restrict__ rstd, float* __restrict__ outT) {
  const int t = blockIdx.x * 256 + threadIdx.x;  // B*N*OCH, channel fastest
  const int o = t & (OCH - 1);
  const int b = t / (NPTS * OCH);
  const float v = (in[t] - mean[b * OCH + o]) * rstd[b * OCH + o];
  outT[t] = v < 0.f ? v * SLOPE : v;
}

__global__ __launch_bounds__(256) void apply_final_kernel(
    const float* __restrict__ y3, const float* __restrict__ mean,
    const float* __restrict__ rstd, float* __restrict__ out) {
  const int t = blockIdx.x * 256 + threadIdx.x;  // B*128*N, n fastest
  const int o = (t >> 13) & 127;
  const int b = t >> 20;
  const float v = (y3[t] - mean[b * 128 + o]) * rstd[b * 128 + o];
  out[t] = v < 0.f ? v * SLOPE : v;
}

// ---------------------------------------------------------------------------
extern "C" void kernel_launch(void* const* d_in, const int* in_sizes, int n_in,
                              void* d_out, int out_size, void* d_ws, size_t ws_size,
                              hipStream_t stream) {
  (void)in_sizes; (void)n_in; (void)out_size; (void)ws_size;
  const float* coords   = (const float*)d_in[0];
  const float* features = (const float*)d_in[1];
  const float* W1 = (const float*)d_in[2];   // [128,256]
  const float* W2 = (const float*)d_in[3];   // [256,256]
  const float* W3 = (const float*)d_in[4];   // [128,512]
  float* out = (float*)d_out;

  float* ws   = (float*)d_ws;
  float* x0T   = ws;                            // B*N*128
  float* x1T   = x0T  + (size_t)BATCH * NPTS * 128;
  float* x2T   = x1T  + (size_t)BATCH * NPTS * 128;   // B*N*256
  float* ymax1 = x2T  + (size_t)BATCH * NPTS * 256;   // B*N*128
  float* ymax2 = ymax1 + (size_t)BATCH * NPTS * 128;  // B*N*256
  float* y3    = ymax2 + (size_t)BATCH * NPTS * 256;  // B*128*N
  float* stats = y3   + (size_t)BATCH * 128 * NPTS;   // 4096 floats
  int*   nidx  = (int*)(stats + 4096);                // B*N*16 ints

  float* ssum1 = stats;        float* ssq1 = stats + 256;
  float* mean1 = stats + 512;  float* rstd1 = stats + 768;
  float* ssum2 = stats + 1024; float* ssq2 = stats + 1536;
  float* mean2 = stats + 2048; float* rstd2 = stats + 2560;
  float* ssum3 = stats + 3072; float* ssq3 = stats + 3328;
  float* mean3 = stats + 3584; float* rstd3 = stats + 3840;

  (void)hipMemsetAsync(stats, 0, 4096 * sizeof(float), stream);

  transpose_feat_kernel<<<(BATCH * NPTS * 128) / 256, 256, 0, stream>>>(features, x0T);
  knn_kernel<<<(BATCH * NPTS) / 256, 256, 0, stream>>>(coords, nidx);

  edge_gemm_kernel<128><<<BATCH * 64, 256, 0, stream>>>(x0T, W1, nidx, ymax1, ssum1, ssq1);
  finalize_stats_kernel<<<1, 256, 0, stream>>>(ssum1, ssq1, mean1, rstd1,
                                               BATCH * 128, 1.f / (float)(NPTS * KNN));
  apply_norm_T_kernel<128><<<(BATCH * NPTS * 128) / 256, 256, 0, stream>>>(ymax1, mean1, rstd1, x1T);

  edge_gemm_kernel<256><<<BATCH * 64, 256, 0, stream>>>(x1T, W2, nidx, ymax2, ssum2, ssq2);
  finalize_stats_kernel<<<2, 256, 0, stream>>>(ssum2, ssq2, mean2, rstd2,
                                               BATCH * 256, 1.f / (float)(NPTS * KNN));
  apply_norm_T_kernel<256><<<(BATCH * NPTS * 256) / 256, 256, 0, stream>>>(ymax2, mean2, rstd2, x2T);

  final_gemm_kernel<<<BATCH * (NPTS / 64), 256, 0, stream>>>(x0T, x1T, x2T, W3, y3, ssum3, ssq3);
  finalize_stats_kernel<<<1, 256, 0, stream>>>(ssum3, ssq3, mean3, rstd3,
                                               BATCH * 128, 1.f / (float)NPTS);
  apply_final_kernel<<<(BATCH * 128 * NPTS) / 256, 256, 0, stream>>>(y3, mean3, rstd3, out);
}